// oup_25881472926368
// MI455X (gfx1250) — compile-verified
//
#include <hip/hip_runtime.h>

// ---------------------------------------------------------------------------
// OU Euler-Maruyama scan, reformulated as blocked linear recurrence:
//   x_{t+1} = a*x_t + b_t,  a = 1-gamma*dt (per n),  b_t = gamma*mu*dt + sigma*sqrt(dt)*eps_t
// 16-step block:  out = L @ B + p * x_in^T   with L[k,j]=a^(k-j) (lower tri), p[k]=a^(k+1)
// -> V_WMMA_F32_16X16X4_F32 chained x4 (full f32 precision).
// Memory-bound: 328 MB min HBM traffic => ~14us @ 23.3 TB/s. Noise (164MB) fits L2,
// so pass-1 reduction + pass-2 WMMA re-read costs ~nothing extra in HBM.
// ---------------------------------------------------------------------------

#define N_PARAM 32
#define M_PATH  64
#define T_STEPS 20000
#define DT      0.01f
#define SQRT_DT 0.1f
#define LC      160                 // chunk length (multiple of 32 and 16)
#define NC      125                 // chunks: 160*125 = 20000
#define NCHAIN  (N_PARAM * M_PATH)  // 2048 independent chains

typedef __attribute__((ext_vector_type(2))) float v2f;
typedef __attribute__((ext_vector_type(8))) float v8f;

__device__ __forceinline__ float pow_int(float a, int e) {
    float r = 1.0f, p = a;
    while (e) { if (e & 1) r *= p; p *= p; e >>= 1; }
    return r;
}

// ---- Kernel 1: per-(chain,chunk) weighted reduction R_c = sum a^(32k+31-lane) * b_{32k+lane} ----
// One wave per (chain, chunk); lane-strided Horner with ratio a^32 -> coalesced reads.
__global__ void k1_chunk_reduce(const float* __restrict__ theta,
                                const float* __restrict__ noise,
                                float* __restrict__ R) {
    int wave = blockIdx.x * (blockDim.x >> 5) + (threadIdx.x >> 5);
    int lane = threadIdx.x & 31;
    int chain = wave / NC;
    int chunk = wave - chain * NC;
    if (chain >= NCHAIN) return;

    int n = chain >> 6;  // chain / 64
    float gamma = theta[n * 4 + 0];
    float mu    = theta[n * 4 + 1];
    float sigma = theta[n * 4 + 2];
    float a   = 1.0f - gamma * DT;
    float cmu = gamma * mu * DT;
    float s   = sigma * SQRT_DT;
    float a32 = pow_int(a, 32);

    const float* src = noise + (long long)chain * T_STEPS + (long long)chunk * LC;

    // r_lane = sum_k b[32k+lane] * a32^k, K = LC/32 = 5
    float r = 0.0f;
#pragma unroll
    for (int k = LC / 32 - 1; k >= 0; --k) {
        float b = cmu + s * src[k * 32 + lane];
        r = r * a32 + b;
    }
    // total weight for this lane: a^(31-lane)
    r *= pow_int(a, 31 - lane);
#pragma unroll
    for (int off = 16; off; off >>= 1)
        r += __shfl_xor(r, off, 32);
    if (lane == 0) R[(long long)chain * NC + chunk] = r;
}

// ---- Kernel 1b: per-chain serial scan over NC chunks -> carry-in x at each chunk ----
__global__ void k1b_carry_scan(const float* __restrict__ theta,
                               const float* __restrict__ R,
                               float* __restrict__ carry) {
    int chain = blockIdx.x * blockDim.x + threadIdx.x;
    if (chain >= NCHAIN) return;
    int n = chain >> 6;
    float a  = 1.0f - theta[n * 4 + 0] * DT;
    float x  = theta[n * 4 + 3];            // x0
    float aL = pow_int(a, LC);
    const float* Rc = R + (long long)chain * NC;
    float* Cc = carry + (long long)chain * NC;
    for (int c = 0; c < NC; ++c) {
        Cc[c] = x;                          // x at start of chunk c (x after step c*LC)
        x = aL * x + Rc[c];
    }
}

// ---- Kernel 2: WMMA f32 16x16x4 blocked scan, 16 time-steps x 16 paths per tile ----
// grid (NC, N_PARAM); block 128 = 4 waves, each wave owns one 16-path tile.
__global__ void k2_wmma_out(const float* __restrict__ theta,
                            const float* __restrict__ noise,
                            const float* __restrict__ carry,
                            float* __restrict__ out) {
    int chunk = blockIdx.x;
    int n     = blockIdx.y;
    int wv    = threadIdx.x >> 5;   // m-tile 0..3
    int lane  = threadIdx.x & 31;
    int col   = lane & 15;
    int hi    = lane >> 4;          // half-wave select

    float gamma = theta[n * 4 + 0];
    float mu    = theta[n * 4 + 1];
    float sigma = theta[n * 4 + 2];
    float a   = 1.0f - gamma * DT;
    float cmu = gamma * mu * DT;
    float s   = sigma * SQRT_DT;

    float pw[17];                   // a^0 .. a^16
    pw[0] = 1.0f;
#pragma unroll
    for (int i = 1; i <= 16; ++i) pw[i] = pw[i - 1] * a;

    // A = L (16x16 lower-triangular of powers of a), 4 slices of 16x4.
    // A-layout (ISA 7.12.2): lane=M (mod 16), VGPR0 holds K=kb, VGPR1 K=kb+1,
    // half-wave hi adds +2 to K.
    v2f A[4];
    int row = col;
#pragma unroll
    for (int sl = 0; sl < 4; ++sl) {
        int kb = 4 * sl + 2 * hi;
        A[sl].x = (kb     <= row) ? pw[row - kb]     : 0.0f;
        A[sl].y = (kb + 1 <= row) ? pw[row - kb - 1] : 0.0f;
    }

    int chain = n * 64 + wv * 16 + col;     // path of this lane's column
    float xin = carry[(long long)chain * NC + chunk];

    const float* src = noise + (long long)chain * T_STEPS + (long long)chunk * LC;
    float*       dst = out   + (long long)chain * T_STEPS + (long long)chunk * LC;

#pragma unroll 1
    for (int blk = 0; blk < LC / 16; ++blk) {
        int tb = blk * 16;
        if (blk + 1 < LC / 16)
            __builtin_prefetch(src + tb + 16, 0, 1);   // global_prefetch_b8 for next tile

        // C = p * x_in^T : C/D layout row = v + 8*hi, col = lane&15
        v8f c;
#pragma unroll
        for (int v = 0; v < 8; ++v)
            c[v] = pw[v + 8 * hi + 1] * xin;

        // 4 chained K-slices: out += L[:,k0:k0+4] @ B[k0:k0+4,:]
#pragma unroll
        for (int sl = 0; sl < 4; ++sl) {
            int kb = 4 * sl + 2 * hi;                  // B rows for this lane
            v2f e = *(const v2f*)(src + tb + kb);      // path=col, t = tb+kb, tb+kb+1
            v2f b;
            b.x = cmu + s * e.x;
            b.y = cmu + s * e.y;
            c = __builtin_amdgcn_wmma_f32_16x16x4_f32(
                    false, A[sl], false, b, (short)0, c, false, false);
        }

        // Store tile: lane writes 8 consecutive t for its path -> two b128 stores
        float* dp = dst + tb + 8 * hi;
        *(float4*)(dp)     = make_float4(c[0], c[1], c[2], c[3]);
        *(float4*)(dp + 4) = make_float4(c[4], c[5], c[6], c[7]);

        // Carry: row 15 (x after 16 steps) lives in c[7] on the hi half-wave.
        xin = __shfl(c[7], 16 + col, 32);
    }
}

extern "C" void kernel_launch(void* const* d_in, const int* in_sizes, int n_in,
                              void* d_out, int out_size, void* d_ws, size_t ws_size,
                              hipStream_t stream) {
    const float* theta = (const float*)d_in[0];
    const float* noise = (const float*)d_in[1];
    float* out = (float*)d_out;

    float* R     = (float*)d_ws;                       // [NCHAIN][NC]
    float* carry = R + (long long)NCHAIN * NC;         // [NCHAIN][NC]

    // K1: 2048*125 = 256000 waves, 8 waves/block -> 32000 blocks
    k1_chunk_reduce<<<(NCHAIN * NC) / 8, 256, 0, stream>>>(theta, noise, R);
    // K1b: one thread per chain
    k1b_carry_scan<<<(NCHAIN + 255) / 256, 256, 0, stream>>>(theta, R, carry);
    // K2: (chunks, n-families), 4 m-tiles per block (one wave each)
    k2_wmma_out<<<dim3(NC, N_PARAM), 128, 0, stream>>>(theta, noise, carry, out);
}